// iCGMMSeq_34737695490691
// MI455X (gfx1250) — compile-verified
//
#include <hip/hip_runtime.h>
#include <stdint.h>

#define NN 200000
#define KK 512
#define CC 64
#define JJ 32

typedef float    v2f  __attribute__((ext_vector_type(2)));
typedef float    v8f  __attribute__((ext_vector_type(8)));
typedef uint32_t u32x4 __attribute__((ext_vector_type(4)));
typedef int      i32x8 __attribute__((ext_vector_type(8)));
typedef int      i32x4 __attribute__((ext_vector_type(4)));

// ---------------------------------------------------------------------------
// TDM: 2D tile load global->LDS. Descriptor per CDNA5 ISA.
// Pads LDS destination: 64-dword interval, 2-dword pad -> 66-dword row stride.
// 6-arg builtin form (amdgpu-toolchain / clang-23).
// ---------------------------------------------------------------------------
__device__ __forceinline__ void tdm_load_2d(uint32_t lds_off, const void* gp,
                                            uint32_t td0, uint32_t td1,
                                            uint32_t tl0, uint32_t tl1,
                                            uint32_t stride0) {
  uint64_t ga = (uint64_t)(uintptr_t)gp;
  u32x4 g0;
  g0.x = 1u;                                              // count=1, user desc
  g0.y = lds_off;                                         // lds_addr [63:32]
  g0.z = (uint32_t)ga;                                    // global_addr lo
  g0.w = ((uint32_t)(ga >> 32) & 0x1FFFFFFu) | (2u << 30); // addr hi + type=2
  uint32_t w0 = (2u << 16)    // data_size = 4 bytes
              | (1u << 20)    // pad_enable
              | (5u << 22)    // pad_interval: 64 dwords
              | (1u << 25);   // pad_amount: 2 dwords  -> stride 66 dwords
  i32x8 g1;
  g1[0] = (int)w0;
  g1[1] = (int)(td0 << 16);                 // tensor_dim0 [79:48]
  g1[2] = (int)((td0 >> 16) | (td1 << 16)); // tensor_dim1 [111:80]
  g1[3] = (int)((td1 >> 16) | (tl0 << 16)); // tile_dim0  [127:112]
  g1[4] = (int)(tl1 & 0xFFFFu);             // tile_dim1 [143:128], tile_dim2=0
  g1[5] = (int)stride0;                     // tensor_dim0_stride [207:160]
  g1[6] = 0;
  g1[7] = 0;                                // tensor_dim1_stride unused (2D)
  i32x4 z4; z4[0] = 0; z4[1] = 0; z4[2] = 0; z4[3] = 0;
  i32x8 z8; z8[0] = 0; z8[1] = 0; z8[2] = 0; z8[3] = 0;
            z8[4] = 0; z8[5] = 0; z8[6] = 0; z8[7] = 0;
  __builtin_amdgcn_tensor_load_to_lds(g0, g1, z4, z4, z8, 0);
}

// ---------------------------------------------------------------------------
// prep1: iv = exp(-log_var); m2 = -2*mu*iv; offs[c] = -0.5*(sum(mu^2*iv)
//        + sum(log_var) + K*ln(2*pi)).  One block per component c.
// ---------------------------------------------------------------------------
__global__ __launch_bounds__(256) void icgmm_prep1(const float* __restrict__ mu,
                                                   const float* __restrict__ lv,
                                                   float* __restrict__ iv,
                                                   float* __restrict__ m2,
                                                   float* __restrict__ offs) {
  __shared__ float red[256];
  const int c = blockIdx.x;
  float t = 0.f;
  for (int k = threadIdx.x; k < KK; k += 256) {
    float l = lv[c * KK + k];
    float m = mu[c * KK + k];
    float v = __expf(-l);
    iv[c * KK + k] = v;
    m2[c * KK + k] = -2.0f * m * v;
    t += m * m * v + l;
  }
  red[threadIdx.x] = t;
  __syncthreads();
  for (int s = 128; s > 0; s >>= 1) {
    if (threadIdx.x < s) red[threadIdx.x] += red[threadIdx.x + s];
    __syncthreads();
  }
  if (threadIdx.x == 0)
    offs[c] = -0.5f * (red[0] + (float)KK * 1.8378770664093453f); // ln(2*pi)
}

// prep2: P[j,c] = log(alpha*beta[c] + njk[j,c]) + offs[c]
__global__ __launch_bounds__(256) void icgmm_prep2(const float* __restrict__ beta,
                                                   const float* __restrict__ njk,
                                                   const float* __restrict__ alpha,
                                                   const float* __restrict__ offs,
                                                   float* __restrict__ P) {
  int i = blockIdx.x * 256 + threadIdx.x;
  if (i < JJ * CC) {
    int c = i & (CC - 1);
    P[i] = logf(alpha[0] * beta[c] + njk[i]) + offs[c];
  }
}

// ---------------------------------------------------------------------------
// main: 128-row tile, full C=64. TDM double-buffers x chunks, single-buffers
// iv/m2 chunks. 8 waves in 4x2 layout, each owns 32x32 via 4 f32 WMMA accums.
// Epilogue: gather P[macro[u]], row softmax, coalesced float4 stores.
// ---------------------------------------------------------------------------
#define XSTR 66   // padded LDS row stride (dwords), matches TDM pad
#define LSTR 65   // logits LDS stride

__global__ __launch_bounds__(256) void icgmm_main(const float* __restrict__ x,
                                                  const float* __restrict__ iv,
                                                  const float* __restrict__ m2,
                                                  const float* __restrict__ P,
                                                  const int* __restrict__ macro,
                                                  float* __restrict__ out) {
  __shared__ float xs[2 * 128 * XSTR];   // 67.6 KB : x tiles, double buffered
  __shared__ float bsv[2 * 64 * XSTR];   // 33.8 KB : iv chunk | m2 chunk
  __shared__ float Pl[JJ * CC];          //  8.0 KB : prior+offset table
  __shared__ float lg[128 * LSTR];       // 33.3 KB : raw quad accumulators
  __shared__ float rmax[128], rinv[128];
  __shared__ int   jr[128];

  const int tid  = threadIdx.x;
  const int wave = tid >> 5;
  const int lane = tid & 31;
  const int half = lane >> 4;      // 0: lanes 0-15, 1: lanes 16-31
  const int l16  = lane & 15;
  const int row0 = blockIdx.x * 128;

  for (int i = tid; i < JJ * CC; i += 256) Pl[i] = P[i];

  const int wr = (wave & 3) * 32;  // wave row offset within tile
  const int wc = (wave >> 2) * 32; // wave col offset within tile

  if (wave == 0) {
    tdm_load_2d((uint32_t)(uintptr_t)&xs[0], x + (size_t)row0 * KK,
                KK, (uint32_t)(NN - row0), 64, 128, KK);
    tdm_load_2d((uint32_t)(uintptr_t)&bsv[0], iv, KK, CC, 64, 64, KK);
    tdm_load_2d((uint32_t)(uintptr_t)&bsv[64 * XSTR], m2, KK, CC, 64, 64, KK);
  }

  v8f acc00 = {0,0,0,0,0,0,0,0};
  v8f acc01 = {0,0,0,0,0,0,0,0};
  v8f acc10 = {0,0,0,0,0,0,0,0};
  v8f acc11 = {0,0,0,0,0,0,0,0};

  for (int kc = 0; kc < 8; ++kc) {
    if (wave == 0) {
      if (kc + 1 < 8) {
        tdm_load_2d((uint32_t)(uintptr_t)&xs[((kc + 1) & 1) * 128 * XSTR],
                    x + (size_t)row0 * KK + (kc + 1) * 64,
                    (uint32_t)(KK - (kc + 1) * 64), (uint32_t)(NN - row0),
                    64, 128, KK);
        __builtin_amdgcn_s_wait_tensorcnt((short)1); // all but newest x done
      } else {
        __builtin_amdgcn_s_wait_tensorcnt((short)0);
      }
    }
    __syncthreads();

    const float* xb = &xs[(kc & 1) * 128 * XSTR];
    const float* bv = &bsv[0];
    const float* bm = &bsv[64 * XSTR];

#pragma unroll 4
    for (int kk = 0; kk < 64; kk += 4) {
      const int ka = kk + 2 * half;  // A/B lane k per 16x16x4 f32 layout
      v2f a0   = *(const v2f*)&xb[(wr + l16) * XSTR + ka];
      v2f a1   = *(const v2f*)&xb[(wr + 16 + l16) * XSTR + ka];
      v2f biv0 = *(const v2f*)&bv[(wc + l16) * XSTR + ka];
      v2f biv1 = *(const v2f*)&bv[(wc + 16 + l16) * XSTR + ka];
      v2f bm0  = *(const v2f*)&bm[(wc + l16) * XSTR + ka];
      v2f bm1  = *(const v2f*)&bm[(wc + 16 + l16) * XSTR + ka];
      v2f a0s = a0 * a0;
      v2f a1s = a1 * a1;
      acc00 = __builtin_amdgcn_wmma_f32_16x16x4_f32(false, a0s, false, biv0, (short)0, acc00, false, false);
      acc01 = __builtin_amdgcn_wmma_f32_16x16x4_f32(false, a0s, false, biv1, (short)0, acc01, false, false);
      acc10 = __builtin_amdgcn_wmma_f32_16x16x4_f32(false, a1s, false, biv0, (short)0, acc10, false, false);
      acc11 = __builtin_amdgcn_wmma_f32_16x16x4_f32(false, a1s, false, biv1, (short)0, acc11, false, false);
      acc00 = __builtin_amdgcn_wmma_f32_16x16x4_f32(false, a0,  false, bm0,  (short)0, acc00, false, false);
      acc01 = __builtin_amdgcn_wmma_f32_16x16x4_f32(false, a0,  false, bm1,  (short)0, acc01, false, false);
      acc10 = __builtin_amdgcn_wmma_f32_16x16x4_f32(false, a1,  false, bm0,  (short)0, acc10, false, false);
      acc11 = __builtin_amdgcn_wmma_f32_16x16x4_f32(false, a1,  false, bm1,  (short)0, acc11, false, false);
    }
    __syncthreads();
    if (wave == 0 && kc + 1 < 8) {  // safe: compute on bsv finished
      tdm_load_2d((uint32_t)(uintptr_t)&bsv[0], iv + (kc + 1) * 64,
                  (uint32_t)(KK - (kc + 1) * 64), CC, 64, 64, KK);
      tdm_load_2d((uint32_t)(uintptr_t)&bsv[64 * XSTR], m2 + (kc + 1) * 64,
                  (uint32_t)(KK - (kc + 1) * 64), CC, 64, 64, KK);
    }
  }

  // pass1: spill accumulators (C/D layout: vgpr v -> row v + 8*half)
#pragma unroll
  for (int v = 0; v < 8; ++v) {
    const int r = wr + v + 8 * half;
    lg[r * LSTR + wc + l16]               = acc00[v];
    lg[r * LSTR + wc + 16 + l16]          = acc01[v];
    lg[(r + 16) * LSTR + wc + l16]        = acc10[v];
    lg[(r + 16) * LSTR + wc + 16 + l16]   = acc11[v];
  }
  __syncthreads();

  // pass2: per-row max & sum of exp(logit), logit = -0.5*quad + P[j,c]
  if (tid < 128) {
    const int u = row0 + tid;
    const int j = (u < NN) ? macro[u] : 0;
    jr[tid] = j;
    const float* Pr = &Pl[j * CC];
    float mx = -3.0e38f;
    for (int c = 0; c < CC; ++c)
      mx = fmaxf(mx, fmaf(-0.5f, lg[tid * LSTR + c], Pr[c]));
    float s = 0.f;
    for (int c = 0; c < CC; ++c)
      s += __expf(fmaf(-0.5f, lg[tid * LSTR + c], Pr[c]) - mx);
    rmax[tid] = mx;
    rinv[tid] = 1.0f / s;
  }
  __syncthreads();

  // pass3: normalized posterior, coalesced float4 stores
#pragma unroll
  for (int it = 0; it < 8; ++it) {
    const int r = (tid >> 4) + 16 * it;
    const int u = row0 + r;
    if (u < NN) {
      const int c4 = (tid & 15) * 4;
      const float* Pr = &Pl[jr[r] * CC];
      const float mx = rmax[r], inv = rinv[r];
      float4 o;
      o.x = __expf(fmaf(-0.5f, lg[r * LSTR + c4 + 0], Pr[c4 + 0]) - mx) * inv;
      o.y = __expf(fmaf(-0.5f, lg[r * LSTR + c4 + 1], Pr[c4 + 1]) - mx) * inv;
      o.z = __expf(fmaf(-0.5f, lg[r * LSTR + c4 + 2], Pr[c4 + 2]) - mx) * inv;
      o.w = __expf(fmaf(-0.5f, lg[r * LSTR + c4 + 3], Pr[c4 + 3]) - mx) * inv;
      *(float4*)&out[(size_t)u * CC + c4] = o;
    }
  }
}

extern "C" void kernel_launch(void* const* d_in, const int* in_sizes, int n_in,
                              void* d_out, int out_size, void* d_ws, size_t ws_size,
                              hipStream_t stream) {
  const float* x     = (const float*)d_in[0];
  const float* mu    = (const float*)d_in[1];
  const float* lv    = (const float*)d_in[2];
  const float* beta  = (const float*)d_in[3];
  const float* njk   = (const float*)d_in[4];
  const float* alpha = (const float*)d_in[5];
  const int*   mac   = (const int*)d_in[6];
  float*       out   = (float*)d_out;

  float* iv   = (float*)d_ws;          // [C*K]
  float* m2   = iv + CC * KK;          // [C*K]
  float* offs = m2 + CC * KK;          // [C]
  float* P    = offs + CC;             // [J*C]

  icgmm_prep1<<<CC, 256, 0, stream>>>(mu, lv, iv, m2, offs);
  icgmm_prep2<<<(JJ * CC + 255) / 256, 256, 0, stream>>>(beta, njk, alpha, offs, P);
  icgmm_main<<<(NN + 127) / 128, 256, 0, stream>>>(x, iv, m2, P, mac, out);
}